// NeuralKNN_11527692223297
// MI455X (gfx1250) — compile-verified
//
#include <hip/hip_runtime.h>

#define B_ 8
#define D_ 128
#define N_ 100000
#define F_ 128
#define K_ 16
#define INV_TEMP 10.0f
#define INV_SQRT_D 0.08838834764831845f   // 1/sqrt(128)

#define NC 64      // n-chunk per block in einsum phase
#define NCP 65     // Wt LDS row stride (odd -> conflict-free strided reads)
#define NCS 68     // Vt LDS row stride: 272B rows = 16B-aligned for async b128,
                   // (4f + j) % 64 bank skew -> conflict-free B-frag reads

typedef __attribute__((ext_vector_type(2))) float v2f;
typedef __attribute__((ext_vector_type(8))) float v8f;
typedef __attribute__((ext_vector_type(4))) int   v4i;

#define AS1 __attribute__((address_space(1)))
#define AS3 __attribute__((address_space(3)))

#ifndef __has_builtin
#define __has_builtin(x) 0
#endif

#if __has_builtin(__builtin_amdgcn_global_load_async_to_lds_b128)
#define HAVE_ASYNC_LDS 1
#else
#define HAVE_ASYNC_LDS 0
#endif

#if HAVE_ASYNC_LDS
#if __has_builtin(__builtin_amdgcn_s_wait_asynccnt)
#define WAIT_ASYNC(n) __builtin_amdgcn_s_wait_asynccnt(n)
#else
#define WAIT_ASYNC(n) asm volatile("s_wait_asynccnt %0" ::"i"(n) : "memory")
#endif
#endif

// ---------------------------------------------------------------------------
// Zero output + reduction scratch (harness poisons them with 0xAA).
// ---------------------------------------------------------------------------
__global__ void nknn_zero_kernel(float* __restrict__ out, float* __restrict__ red) {
    int i = blockIdx.x * blockDim.x + threadIdx.x;
    if (i < B_ * K_ * F_) out[i] = 0.0f;
    if (i < (K_ + 1) * B_) red[i] = 0.0f;
}

// ---------------------------------------------------------------------------
// sims[b,n] = -(q[b,:] . keys[b,:,n]) / sqrt(D); red[b] += sum exp(sims)
// keys streamed once (410 MB), NT hint: no reuse.
// ---------------------------------------------------------------------------
__global__ void nknn_sims_kernel(const float* __restrict__ q,
                                 const float* __restrict__ keys,
                                 float* __restrict__ sims,
                                 float* __restrict__ red) {
    const int b = blockIdx.y;
    __shared__ float qs[D_];
    __shared__ float bsum;
    if (threadIdx.x < D_) qs[threadIdx.x] = q[b * D_ + threadIdx.x];
    if (threadIdx.x == 0) bsum = 0.0f;
    __syncthreads();

    const float* kb = keys + (size_t)b * D_ * N_;
    float lsum = 0.0f;
    for (int n = blockIdx.x * blockDim.x + threadIdx.x; n < N_;
         n += gridDim.x * blockDim.x) {
        float acc = 0.0f;
#pragma unroll 8
        for (int d = 0; d < D_; ++d)
            acc = fmaf(qs[d], __builtin_nontemporal_load(&kb[(size_t)d * N_ + n]), acc);
        float s = -acc * INV_SQRT_D;
        sims[(size_t)b * N_ + n] = s;
        lsum += __expf(s);
    }
    atomicAdd(&bsum, lsum);
    __syncthreads();
    if (threadIdx.x == 0) atomicAdd(&red[b], bsum);
}

// ---------------------------------------------------------------------------
// alpha[b,n] = exp(sims)/Z0 ; red[B + b] += sum exp(alpha/T)
// ---------------------------------------------------------------------------
__global__ void nknn_alpha0_kernel(const float* __restrict__ sims,
                                   float* __restrict__ alpha,
                                   float* __restrict__ red) {
    const int b = blockIdx.y;
    __shared__ float bsum;
    if (threadIdx.x == 0) bsum = 0.0f;
    __syncthreads();
    const float invZ = 1.0f / red[b];
    float lsum = 0.0f;
    for (int n = blockIdx.x * blockDim.x + threadIdx.x; n < N_;
         n += gridDim.x * blockDim.x) {
        float a = __expf(sims[(size_t)b * N_ + n]) * invZ;
        alpha[(size_t)b * N_ + n] = a;
        lsum += __expf(INV_TEMP * a);
    }
    atomicAdd(&bsum, lsum);
    __syncthreads();
    if (threadIdx.x == 0) atomicAdd(&red[B_ + b], bsum);
}

// ---------------------------------------------------------------------------
// One soft-KNN round: w = exp(alpha/T)/S_k ; alpha += log1p(-w);
// accumulate next round's S into red[(2+k)*B + b].
// ---------------------------------------------------------------------------
__global__ void nknn_iter_kernel(float* __restrict__ alpha,
                                 float* __restrict__ red, int k) {
    const int b = blockIdx.y;
    __shared__ float bsum;
    if (threadIdx.x == 0) bsum = 0.0f;
    __syncthreads();
    const float invS = 1.0f / red[(1 + k) * B_ + b];
    float lsum = 0.0f;
    for (int n = blockIdx.x * blockDim.x + threadIdx.x; n < N_;
         n += gridDim.x * blockDim.x) {
        float a = alpha[(size_t)b * N_ + n];
        float w = __expf(INV_TEMP * a) * invS;
        a += log1pf(-w);
        alpha[(size_t)b * N_ + n] = a;
        if (k + 1 < K_) lsum += __expf(INV_TEMP * a);
    }
    if (k + 1 < K_) {
        atomicAdd(&bsum, lsum);
        __syncthreads();
        if (threadIdx.x == 0) atomicAdd(&red[(2 + k) * B_ + b], bsum);
    }
}

// ---------------------------------------------------------------------------
// Helpers for the WMMA einsum phase.
// ---------------------------------------------------------------------------
__device__ __forceinline__ void recompute_W(const float* __restrict__ sb,
                                            const float* __restrict__ Sks,
                                            float (*Wt)[NCP], int n0, int tid) {
    if (tid < NC) {
        int n = n0 + tid;
        if (n < N_) {
            float a = __expf(sb[n]) / Sks[0];
#pragma unroll
            for (int k = 0; k < K_; ++k) {
                float w = __expf(INV_TEMP * a) / Sks[k + 1];
                Wt[k][tid] = w;
                a += log1pf(-w);
            }
        } else {
#pragma unroll
            for (int k = 0; k < K_; ++k) Wt[k][tid] = 0.0f;
        }
    }
}

// WMMA: A = W[16 x 4], B = V^T[4 x 16], D += A*B (ISA 7.12.2 layouts:
// lanes0-15 hold K=0,1 in v0,v1; lanes16-31 hold K=2,3).
__device__ __forceinline__ v8f wmma_tiles(v8f acc, const float (*Vtb)[NCS],
                                          const float (*Wt)[NCP],
                                          int lane, int wave) {
    const int row = lane & 15;
    const int kb2 = (lane >> 4) << 1;
    const int f0  = wave << 4;
#pragma unroll
    for (int j0 = 0; j0 < NC; j0 += 4) {
        v2f afr, bfr;
        afr.x = Wt[row][j0 + kb2];
        afr.y = Wt[row][j0 + kb2 + 1];
        bfr.x = Vtb[f0 + row][j0 + kb2];
        bfr.y = Vtb[f0 + row][j0 + kb2 + 1];
        acc = __builtin_amdgcn_wmma_f32_16x16x4_f32(
            false, afr, false, bfr, (short)0, acc, false, false);
    }
    return acc;
}

#if HAVE_ASYNC_LDS
// Issue one 128 x NC f32 tile as 2048 b128 async DMA segments (8 per thread,
// so per-wave ASYNCcnt rises by exactly 8 per chunk).
__device__ __forceinline__ void issue_async_tile(const float* __restrict__ vb,
                                                 int n0, float (*Vtb)[NCS],
                                                 int tid) {
#pragma unroll
    for (int it = 0; it < 8; ++it) {
        int seg = tid + it * 256;       // 0..2047
        int f = seg >> 4;               // 16 segments per row
        int s = seg & 15;
        AS1 v4i* src = (AS1 v4i*)(vb + (size_t)f * N_ + n0 + s * 4);
        AS3 v4i* dst = (AS3 v4i*)&Vtb[f][s * 4];
        __builtin_amdgcn_global_load_async_to_lds_b128(src, dst, 0, 0);
    }
}
#endif

// ---------------------------------------------------------------------------
// out[b,k,f] += sum_n w_k(n) * values[b,f,n]  via V_WMMA_F32_16X16X4_F32.
// Values tile staged in LDS with double-buffered GLOBAL_LOAD_ASYNC_TO_LDS
// (ASYNCcnt-fenced) when available; W tile recomputed from sims + the 17
// precomputed softmax scalars. 8 waves each own a 16-wide f-tile and keep the
// f32 D-fragment resident across chunks; one atomic f32 add per element at end.
// ---------------------------------------------------------------------------
__global__ void nknn_einsum_kernel(const float* __restrict__ values,
                                   const float* __restrict__ sims,
                                   const float* __restrict__ red,
                                   float* __restrict__ out) {
    const int b = blockIdx.y;
    __shared__ float Vt[2][F_][NCS];   // double-buffered values tile
    __shared__ float Wt[K_][NCP];      // soft-KNN weights tile
    __shared__ float Sks[K_ + 1];      // Z0, S_1..S_16

    const int tid  = threadIdx.x;
    const int wave = tid >> 5;
    const int lane = tid & 31;

    if (tid <= K_) Sks[tid] = red[tid * B_ + b];
    __syncthreads();

    const float* vb = values + (size_t)b * F_ * N_;
    const float* sb = sims + (size_t)b * N_;

    v8f acc = {};

#if HAVE_ASYNC_LDS
    const int G = gridDim.x;
    const int nFull = N_ / NC;         // full chunks only in the async loop
    int c = blockIdx.x;
    int p = 0;
    if (c < nFull) issue_async_tile(vb, c * NC, Vt[0], tid);
    for (; c < nFull; c += G, p ^= 1) {
        const int cn = c + G;
        if (cn < nFull) issue_async_tile(vb, cn * NC, Vt[p ^ 1], tid);
        recompute_W(sb, Sks, Wt, c * NC, tid);
        if (cn < nFull) { WAIT_ASYNC(8); } else { WAIT_ASYNC(0); }
        __syncthreads();               // Vt[p] + Wt visible to all waves
        acc = wmma_tiles(acc, Vt[p], Wt, lane, wave);
        __syncthreads();               // protect Wt / Vt[p] before reuse
    }
    // Tail chunk (N % NC): guarded synchronous path, one block only.
    if (blockIdx.x == 0 && nFull * NC < N_) {
        const int n0 = nFull * NC;
        for (int idx = tid; idx < F_ * NC; idx += 256) {
            int f = idx / NC, j = idx - f * NC;
            int n = n0 + j;
            Vt[0][f][j] = (n < N_)
                ? __builtin_nontemporal_load(&vb[(size_t)f * N_ + n]) : 0.0f;
        }
        recompute_W(sb, Sks, Wt, n0, tid);
        __syncthreads();
        acc = wmma_tiles(acc, Vt[0], Wt, lane, wave);
    }
#else
    // Fallback: synchronous single-buffer staging over all chunks.
    const int nChunks = (N_ + NC - 1) / NC;
    for (int c = blockIdx.x; c < nChunks; c += gridDim.x) {
        const int n0 = c * NC;
        for (int idx = tid; idx < F_ * NC; idx += 256) {
            int f = idx / NC, j = idx - f * NC;
            int n = n0 + j;
            Vt[0][f][j] = (n < N_)
                ? __builtin_nontemporal_load(&vb[(size_t)f * N_ + n]) : 0.0f;
        }
        recompute_W(sb, Sks, Wt, n0, tid);
        __syncthreads();
        acc = wmma_tiles(acc, Vt[0], Wt, lane, wave);
        __syncthreads();
    }
#endif

    // D frag: VGPR r -> row M=r (lanes0-15) / M=r+8 (lanes16-31); col = lane&15.
    const int kbase = (lane >> 4) << 3;
    const int f = (wave << 4) + (lane & 15);
#pragma unroll
    for (int r = 0; r < 8; ++r) {
        int k = r + kbase;
        atomicAdd(&out[((size_t)b * K_ + k) * F_ + f], acc[r]);
    }
}

// ---------------------------------------------------------------------------
extern "C" void kernel_launch(void* const* d_in, const int* in_sizes, int n_in,
                              void* d_out, int out_size, void* d_ws, size_t ws_size,
                              hipStream_t stream) {
    (void)in_sizes; (void)n_in; (void)out_size; (void)ws_size;
    const float* query  = (const float*)d_in[0];   // [B,D]
    const float* keys   = (const float*)d_in[1];   // [B,D,N]
    const float* values = (const float*)d_in[2];   // [B,F,N]
    float* out = (float*)d_out;                    // [B,K,F]

    float* sims  = (float*)d_ws;                   // B*N
    float* alpha = sims + (size_t)B_ * N_;         // B*N
    float* red   = alpha + (size_t)B_ * N_;        // (K+1)*B

    nknn_zero_kernel<<<(B_ * K_ * F_ + 255) / 256, 256, 0, stream>>>(out, red);

    nknn_sims_kernel<<<dim3(200, B_), 256, 0, stream>>>(query, keys, sims, red);

    nknn_alpha0_kernel<<<dim3(64, B_), 256, 0, stream>>>(sims, alpha, red);

    for (int k = 0; k < K_; ++k)
        nknn_iter_kernel<<<dim3(64, B_), 256, 0, stream>>>(alpha, red, k);

    nknn_einsum_kernel<<<dim3(160, B_), 256, 0, stream>>>(values, sims, red, out);
}